// SqueezeExcitation_12610023981151
// MI455X (gfx1250) — compile-verified
//
#include <hip/hip_runtime.h>
#include <math.h>

typedef __attribute__((ext_vector_type(2))) float v2f;
typedef __attribute__((ext_vector_type(4))) float v4f;
typedef __attribute__((ext_vector_type(8))) float v8f;

#define C_IN  256
#define C_BN  16
#define BATCH 64
#define HW    4096   // 64*64 spatial elements per (b,c) plane

// ---------------------------------------------------------------------------
// Kernel 1: global average pool. One 256-thread block per (b,c) plane.
// Each thread streams 4 float4 (16 floats) -> LDS tree reduce -> pooled[bc].
// ---------------------------------------------------------------------------
__global__ __launch_bounds__(256) void se_pool(const float* __restrict__ x,
                                               float* __restrict__ pooled) {
  __shared__ float red[256];
  const int bc = blockIdx.x;
  const v4f* xp = (const v4f*)(x + (size_t)bc * HW);
  const int t = threadIdx.x;

  float s = 0.0f;
#pragma unroll
  for (int i = 0; i < 4; ++i) {
    v4f v = xp[t + i * 256];
    s += v.x + v.y + v.z + v.w;
  }
  red[t] = s;
  __syncthreads();
#pragma unroll
  for (int off = 128; off > 0; off >>= 1) {
    if (t < off) red[t] += red[t + off];
    __syncthreads();
  }
  if (t == 0) pooled[bc] = red[0] * (1.0f / (float)HW);
}

// ---------------------------------------------------------------------------
// Kernel 2: SE MLP via V_WMMA_F32_16X16X4_F32 (full f32 precision).
//   H     = GELU( pooled[64,256] @ w1^T[256,16] )      w1: [16,256]
//   scale = sigmoid( H[64,16]    @ w2^T[16,256]  )     w2: [256,16]
// One block of 128 threads = 4 waves; wave w owns rows [16w, 16w+16).
//
// A(16x4) lane layout (ISA 7.12.2): lane l, hi=l>=16, m=l&15:
//   VGPR0 = A[m][k0+2*hi], VGPR1 = A[m][k0+2*hi+1]   -> contiguous float2
// B(4x16) symmetric: lane n=l&15: VGPR0 = B[k0+2*hi][n], VGPR1 = B[k0+2*hi+1][n]
// D(16x16): reg r, lanes0-15 -> M=r, lanes16-31 -> M=r+8, N = l&15
// ---------------------------------------------------------------------------
__global__ __launch_bounds__(128) void se_mlp(const float* __restrict__ pooled,
                                              const float* __restrict__ w1,
                                              const float* __restrict__ w2,
                                              float* __restrict__ scale) {
  __shared__ float Hlds[4][16][16];

  const int lane = threadIdx.x & 31;
  const int wave = threadIdx.x >> 5;
  const int hi   = lane >> 4;     // 0: lanes 0-15, 1: lanes 16-31
  const int ln   = lane & 15;
  const int row0 = wave * 16;

  // ---- GEMM1: 16x16 tile of H, K=256 in 64 steps of 4 ----
  v8f acc = {};
#pragma unroll 8
  for (int k0 = 0; k0 < C_IN; k0 += 4) {
    // A: pooled rows (M-tile), K slice
    v2f a = *(const v2f*)(pooled + (size_t)(row0 + ln) * C_IN + k0 + 2 * hi);
    // B[k][n] = w1^T[k][n] = w1[n][k]  (w1 row-major [16][256])
    v2f b = *(const v2f*)(w1 + (size_t)ln * C_IN + k0 + 2 * hi);
    acc = __builtin_amdgcn_wmma_f32_16x16x4_f32(false, a, false, b,
                                                (short)0, acc, false, false);
  }

  // Exact GELU on D-layout, stage to LDS in row-major [m][n]
#pragma unroll
  for (int r = 0; r < 8; ++r) {
    float h = acc[r];
    float g = 0.5f * h * (1.0f + erff(h * 0.70710678118654752f));
    Hlds[wave][r + 8 * hi][ln] = g;
  }
  __syncthreads();

  // ---- GEMM2: scale tile = H(16x16) @ w2^T(16x256), 16 N-tiles ----
  for (int nt = 0; nt < 16; ++nt) {
    v8f acc2 = {};
#pragma unroll
    for (int kk = 0; kk < 4; ++kk) {
      const int k0 = kk * 4;
      // A = H slice, re-read from LDS in A layout (8B-aligned float2)
      v2f a = *(const v2f*)&Hlds[wave][ln][k0 + 2 * hi];
      // B[k][n] = w2^T[k][nt*16+n] = w2[nt*16+n][k]  (w2 row-major [256][16])
      v2f b = *(const v2f*)(w2 + (size_t)(nt * 16 + ln) * C_BN + k0 + 2 * hi);
      acc2 = __builtin_amdgcn_wmma_f32_16x16x4_f32(false, a, false, b,
                                                   (short)0, acc2, false, false);
    }
#pragma unroll
    for (int r = 0; r < 8; ++r) {
      float v  = acc2[r];
      float sg = 1.0f / (1.0f + expf(-v));
      scale[(size_t)(row0 + r + 8 * hi) * C_IN + nt * 16 + ln] = sg;
    }
  }
}

// ---------------------------------------------------------------------------
// Kernel 3: out = x * scale[bc], one block per (b,c) plane.
// scale is uniform per block (scalar/broadcast load). Non-temporal stores for
// `out` so the 256MB output stream does not evict x from the 192MB L2.
// ---------------------------------------------------------------------------
__global__ __launch_bounds__(256) void se_apply(const float* __restrict__ x,
                                                const float* __restrict__ scale,
                                                float* __restrict__ out) {
  const int bc = blockIdx.x;
  const float s = scale[bc];
  const v4f* xp = (const v4f*)(x + (size_t)bc * HW);
  v4f* op = (v4f*)(out + (size_t)bc * HW);
  const int t = threadIdx.x;
#pragma unroll
  for (int i = 0; i < 4; ++i) {
    v4f v = xp[t + i * 256];
    v *= s;
    __builtin_nontemporal_store(v, &op[t + i * 256]);
  }
}

// ---------------------------------------------------------------------------
extern "C" void kernel_launch(void* const* d_in, const int* in_sizes, int n_in,
                              void* d_out, int out_size, void* d_ws, size_t ws_size,
                              hipStream_t stream) {
  const float* x  = (const float*)d_in[0];   // [64,256,64,64]
  const float* w1 = (const float*)d_in[1];   // [16,256]
  const float* w2 = (const float*)d_in[2];   // [256,16]
  float* out = (float*)d_out;

  float* pooled = (float*)d_ws;              // 64*256 floats (64 KB)
  float* scale  = pooled + BATCH * C_IN;     // 64*256 floats (64 KB)

  se_pool <<<BATCH * C_IN, 256, 0, stream>>>(x, pooled);
  se_mlp  <<<1,            128, 0, stream>>>(pooled, w1, w2, scale);
  se_apply<<<BATCH * C_IN, 256, 0, stream>>>(x, scale, out);
}